// PyTorchProjectorModule_23957327577513
// MI455X (gfx1250) — compile-verified
//
#include <hip/hip_runtime.h>
#include <stdint.h>
#include <math.h>

// ---------------------------------------------------------------------------
// 8-view projector over a 128^3 volume:
//   out[n,j,k] = scale * sum_t sum_{4 xy-corners} w(n,j,t,c) * vol[cell*128 + k]
// (z-interpolation is exact: iz == k, so the trilinear gather degenerates to a
//  bilinear xy gather of full contiguous 512B z-rows.)
//
// One block (2 wave32) per (n,j) ray. Each wave owns half of the 56 depth
// chunks and runs a 3-deep async pipeline: the CDNA5 Tensor Data Mover
// gather-mode pulls 16 x 512B volume rows per chunk into LDS (TENSORcnt),
// while the wave FMAs the chunk that already landed. Partials are combined
// in a fixed order -> deterministic output.
// ---------------------------------------------------------------------------

typedef __attribute__((ext_vector_type(4))) unsigned int v4u_t;
typedef __attribute__((ext_vector_type(8))) int          v8i_t;
typedef __attribute__((ext_vector_type(4))) int          v4i_t;

#define NDEPTH 222      // ceil(sqrt(3)*128)
#define TPAD   224      // padded to multiple of 4 (zero-weight tail)
#define NCHUNK (TPAD/4) // 56 gather chunks of 4 steps (16 rows) each
#define CPW    (NCHUNK/2) // 28 chunks per wave
#define PDEPTH 3        // per-wave TDM pipeline depth

// Issue one gather-mode TDM load: 16 rows x 128 f32 (8KB) into LDS at byte address lds_off.
// D# layout per CDNA5 ISA 8.3-8.7:
//   g0: count=1, gather_mode=1, 16-bit indices, lds_addr, global_addr, type=2
//   g1: data_size=4B, tensor_dim0=128, tensor_dim1=16384, tile_dim0=128,
//       tile_dim1=16 (#indices), tensor_dim0_stride=128
//   g2/g3: 16 packed 16-bit row indices
__device__ __forceinline__ void tdm_gather16(const float* vol, uint32_t lds_off,
                                             v4i_t g2, v4i_t g3)
{
  const uint64_t ga = (uint64_t)(uintptr_t)vol;
  v4u_t g0;
  g0[0] = 1u | (1u << 31);                                    // count=1 | gather_mode
  g0[1] = lds_off;                                            // LDS byte address
  g0[2] = (uint32_t)ga;                                       // global_addr[31:0]
  g0[3] = (uint32_t)((ga >> 32) & 0x1FFFFFFull) | (2u << 30); // addr[56:32] | type=2
  v8i_t g1;
  g1[0] = (int)(2u << 16);      // data_size = 4 bytes (code 2)
  g1[1] = (int)(128u << 16);    // tensor_dim0 = 128 (bits 79:48, low half)
  g1[2] = (int)(16384u << 16);  // tensor_dim1 = 16384 rows (bits 111:80, low half)
  g1[3] = (int)(128u << 16);    // tile_dim0 = 128 (bits 127:112)
  g1[4] = 16;                   // tile_dim1 = 16 valid gather indices
  g1[5] = 128;                  // tensor_dim0_stride = 128 elements
  g1[6] = 0;
  g1[7] = 0;
#if defined(__clang_major__) && (__clang_major__ >= 23)
  v8i_t g4 = {0, 0, 0, 0, 0, 0, 0, 0};   // clang-23 toolchain: 6-arg form
  __builtin_amdgcn_tensor_load_to_lds(g0, g1, g2, g3, g4, 0);
#else
  __builtin_amdgcn_tensor_load_to_lds(g0, g1, g2, g3, 0);  // ROCm 7.2 5-arg form
#endif
}

__global__ __launch_bounds__(64) void proj_tdm_kernel(
    const float* __restrict__ vol,   // 128*128*128, layout (x*128+y)*128+z
    const float* __restrict__ vec,   // 8 x 12: [ray(3), dc(3), u(3), v(3)]
    float* __restrict__ out,         // 8 x 128(j) x 128(k)
    float neg_me, float t_step, float scale)
{
  __shared__ __align__(16) uint32_t idx_tab[TPAD * 2];            // packed u16 cells
  __shared__ __align__(16) float    w_tab[TPAD * 4];              // corner weights
  __shared__ __align__(16) float    rowbuf[2][PDEPTH][16 * 128];  // gather tiles
  __shared__ __align__(16) float4   part[32];                     // wave1 partials

  const int tid  = threadIdx.x;
  const int wid  = tid >> 5;      // wave 0 / wave 1
  const int lane = tid & 31;      // owns k = 4*lane .. 4*lane+3
  const int j    = blockIdx.x;
  const int n    = blockIdx.y;

  const float rx  = vec[n * 12 + 0], ry  = vec[n * 12 + 1];
  const float dcx = vec[n * 12 + 3], dcy = vec[n * 12 + 4];
  const float ux  = vec[n * 12 + 6], uy  = vec[n * 12 + 7];
  const float jc  = (float)j - 63.5f;
  const float bx  = dcx + jc * ux;
  const float by  = dcy + jc * uy;

  // ---- Phase 1: per-step bilinear corner cells + weights (z exact -> 2D) ----
  for (int t = tid; t < TPAD; t += 64) {
    float w0 = 0.f, w1 = 0.f, w2 = 0.f, w3 = 0.f;
    uint32_t c0 = 0, c1 = 0, c2 = 0, c3 = 0;
    if (t < NDEPTH) {
      const float tc  = fmaf((float)t, t_step, neg_me);
      const float px  = bx + tc * rx;
      const float py  = by + tc * ry;
      // grid_sample mapping, align_corners=False: ix = ((p/64 + 1)*128 - 1)/2
      const float ixf = ((px * (1.f / 64.f) + 1.f) * 128.f - 1.f) * 0.5f;
      const float iyf = ((py * (1.f / 64.f) + 1.f) * 128.f - 1.f) * 0.5f;
      const float xf0 = floorf(ixf), yf0 = floorf(iyf);
      const float fx  = ixf - xf0,  fy  = iyf - yf0;
      const int x0 = (int)xf0, y0 = (int)yf0;
      const int x1 = x0 + 1,   y1 = y0 + 1;
      const bool vx0 = (x0 >= 0) & (x0 < 128), vx1 = (x1 >= 0) & (x1 < 128);
      const bool vy0 = (y0 >= 0) & (y0 < 128), vy1 = (y1 >= 0) & (y1 < 128);
      const int x0c = min(max(x0, 0), 127), x1c = min(max(x1, 0), 127);
      const int y0c = min(max(y0, 0), 127), y1c = min(max(y1, 0), 127);
      w0 = (1.f - fx) * (1.f - fy) * ((vx0 & vy0) ? 1.f : 0.f);
      w1 = (1.f - fx) * fy         * ((vx0 & vy1) ? 1.f : 0.f);
      w2 = fx * (1.f - fy)         * ((vx1 & vy0) ? 1.f : 0.f);
      w3 = fx * fy                 * ((vx1 & vy1) ? 1.f : 0.f);
      c0 = (uint32_t)(x0c * 128 + y0c);
      c1 = (uint32_t)(x0c * 128 + y1c);
      c2 = (uint32_t)(x1c * 128 + y0c);
      c3 = (uint32_t)(x1c * 128 + y1c);
    }
    idx_tab[t * 2 + 0] = c0 | (c1 << 16);
    idx_tab[t * 2 + 1] = c2 | (c3 << 16);
    w_tab[t * 4 + 0] = w0;  w_tab[t * 4 + 1] = w1;
    w_tab[t * 4 + 2] = w2;  w_tab[t * 4 + 3] = w3;
  }
  __syncthreads();   // tables complete before either wave's TDM reads them

  // Broadcast-load 16 packed indices for chunk q into scalar descriptor groups.
  auto ld_g23 = [&](int q, v4i_t& g2, v4i_t& g3) {
    const uint4 a = *(const uint4*)&idx_tab[q * 8];
    const uint4 b = *(const uint4*)&idx_tab[q * 8 + 4];
    g2[0] = __builtin_amdgcn_readfirstlane((int)a.x);
    g2[1] = __builtin_amdgcn_readfirstlane((int)a.y);
    g2[2] = __builtin_amdgcn_readfirstlane((int)a.z);
    g2[3] = __builtin_amdgcn_readfirstlane((int)a.w);
    g3[0] = __builtin_amdgcn_readfirstlane((int)b.x);
    g3[1] = __builtin_amdgcn_readfirstlane((int)b.y);
    g3[2] = __builtin_amdgcn_readfirstlane((int)b.z);
    g3[3] = __builtin_amdgcn_readfirstlane((int)b.w);
  };

  // LDS byte offsets (flat-address low 32 bits == LDS offset per aperture calc)
  uint32_t boff[PDEPTH];
#pragma unroll
  for (int b = 0; b < PDEPTH; ++b)
    boff[b] = (uint32_t)(uintptr_t)&rowbuf[wid][b][0];

  // ---- Phase 2: 3-deep TDM gather + FMA pipeline, CPW chunks per wave ----
  const int cbase = wid * CPW;
#pragma unroll
  for (int p = 0; p < PDEPTH; ++p) {
    v4i_t g2, g3; ld_g23(cbase + p, g2, g3);
    tdm_gather16(vol, boff[p], g2, g3);
  }

  float4 acc = make_float4(0.f, 0.f, 0.f, 0.f);
  const float4* wt4 = (const float4*)w_tab;

  for (int i = 0; i < CPW; ++i) {
    // Wait until chunk i has landed: allow min(PDEPTH-1, CPW-1-i) in flight.
    if (i + 2 < CPW)      __builtin_amdgcn_s_wait_tensorcnt(2);
    else if (i + 1 < CPW) __builtin_amdgcn_s_wait_tensorcnt(1);
    else                  __builtin_amdgcn_s_wait_tensorcnt(0);
    __builtin_amdgcn_sched_barrier(0);

    const int cc = cbase + i;
    const float4* rb = (const float4*)&rowbuf[wid][i % PDEPTH][0];
#pragma unroll
    for (int s = 0; s < 16; ++s) {          // 16 gathered rows this chunk
      const float4 wv = wt4[cc * 4 + (s >> 2)];           // broadcast ds_load_b128
      const float ws = ((s & 3) == 0) ? wv.x :
                       ((s & 3) == 1) ? wv.y :
                       ((s & 3) == 2) ? wv.z : wv.w;
      const float4 r = rb[s * 32 + lane];                 // ds_load_b128, 4 k's
      acc.x = fmaf(ws, r.x, acc.x);
      acc.y = fmaf(ws, r.y, acc.y);
      acc.z = fmaf(ws, r.z, acc.z);
      acc.w = fmaf(ws, r.w, acc.w);
    }
    if (i + PDEPTH < CPW) {                 // refill the buffer just consumed
      v4i_t g2, g3; ld_g23(cc + PDEPTH, g2, g3);
      tdm_gather16(vol, boff[i % PDEPTH], g2, g3);
    }
  }

  // ---- Fixed-order cross-wave combine (deterministic) + store ----
  if (wid == 1) part[lane] = acc;
  __syncthreads();
  if (wid == 0) {
    const float4 p = part[lane];
    float4 o;
    o.x = (acc.x + p.x) * scale;
    o.y = (acc.y + p.y) * scale;
    o.z = (acc.z + p.z) * scale;
    o.w = (acc.w + p.w) * scale;
    *(float4*)(out + ((size_t)n * 128 + j) * 128 + (size_t)lane * 4) = o;
  }
}

extern "C" void kernel_launch(void* const* d_in, const int* in_sizes, int n_in,
                              void* d_out, int out_size, void* d_ws, size_t ws_size,
                              hipStream_t stream) {
  (void)in_sizes; (void)n_in; (void)out_size; (void)d_ws; (void)ws_size;
  const float* vol = (const float*)d_in[0];   // (128,128,128) f32
  const float* vec = (const float*)d_in[1];   // (8,12) f32
  float* out = (float*)d_out;                 // (8,128,128) f32

  const double me    = 96.0 * sqrt(3.0);      // sqrt(3*64^2)*1.5
  const float neg_me = (float)(-me);
  const float t_step = (float)(2.0 * me / (double)(NDEPTH - 1));
  const float scale  = (float)(2.0 * me / (double)NDEPTH);

  dim3 grid(128, 8, 1);   // (j, n): one 2-wave block per ray
  dim3 block(64, 1, 1);
  proj_tdm_kernel<<<grid, block, 0, stream>>>(vol, vec, out,
                                              neg_me, t_step, scale);
}